// BinaryMatchingPursuit_66503273611581
// MI455X (gfx1250) — compile-verified
//
#include <hip/hip_runtime.h>
#include <hip/hip_bf16.h>

#define BATCH 512
#define NIN   512
#define NOUT  2048
#define KACT  21   // ceil(0.01 * 2048)
#define KIN   6    // ceil(0.01 * 512)

typedef __attribute__((ext_vector_type(2))) float v2f;
typedef __attribute__((ext_vector_type(8))) float v8f;

#define WMMA_F32(a, b, c) \
    __builtin_amdgcn_wmma_f32_16x16x4_f32(false, (a), false, (b), (short)0, (c), false, false)

// ---------------------------------------------------------------------------
// lambd = 3 * max_b(nonzero(x[b,:])) * NIN
// ---------------------------------------------------------------------------
__global__ __launch_bounds__(512) void lambd_kernel(const float* __restrict__ x,
                                                    float* __restrict__ lam) {
    __shared__ int smax[512];
    const int b = threadIdx.x;
    int cnt = 0;
    for (int i = 0; i < NIN; ++i)
        cnt += (x[b * NIN + i] != 0.0f) ? 1 : 0;
    smax[b] = cnt;
    __syncthreads();
    for (int s = 256; s > 0; s >>= 1) {
        if (b < s) smax[b] = max(smax[b], smax[b + s]);
        __syncthreads();
    }
    if (b == 0) lam[0] = 3.0f * (float)smax[0] * (float)NIN;
}

// ---------------------------------------------------------------------------
// Wt[i, o] = W[o, i]   (column-major copy for coalesced column sums)
// ---------------------------------------------------------------------------
__global__ __launch_bounds__(256) void transpose_w(const float* __restrict__ w,
                                                   float* __restrict__ wt) {
    __shared__ float tile[32][33];
    const int bx = blockIdx.x * 32;          // along NIN
    const int by = blockIdx.y * 32;          // along NOUT
    const int tx = threadIdx.x & 31;
    const int ty = threadIdx.x >> 5;         // 0..7
    #pragma unroll
    for (int r = 0; r < 32; r += 8)
        tile[ty + r][tx] = w[(by + ty + r) * NIN + bx + tx];
    __syncthreads();
    #pragma unroll
    for (int r = 0; r < 32; r += 8)
        wt[(bx + ty + r) * NOUT + by + tx] = tile[tx][ty + r];
}

// ---------------------------------------------------------------------------
// G0 = x @ W^T via V_WMMA_F32_16X16X4_F32 with 2x2 register blocking.
// Each wave: 32(M) x 32(N) macro-tile = 4 accumulators; per K-step loads
// 2 A-fragments + 2 B-fragments (b64 each) and issues 4 WMMAs (1.0 load/WMMA).
// Block = 8 waves arranged 2(M) x 4(N) -> 64 x 128 per block.
// A 16x4 f32 layout: lane&15 -> M row, K = (lane>>4)*2 + {0,1} in the VGPR pair.
// B 4x16 f32 layout: lane&15 -> N col (= W row), same K striping.
// C 16x16 f32: element (M = v + 8*(lane>>4), N = lane&15) in c[v].
// ---------------------------------------------------------------------------
__global__ __launch_bounds__(256) void gemm_xwT(const float* __restrict__ x,
                                                const float* __restrict__ w,
                                                float* __restrict__ g0) {
    const int lane = threadIdx.x & 31;
    const int wave = threadIdx.x >> 5;
    const int half = lane >> 4;
    const int sub  = lane & 15;
    const int m0 = blockIdx.y * 64  + (wave >> 2) * 32;
    const int n0 = blockIdx.x * 128 + (wave & 3) * 32;

    const float* a0p = x + (size_t)(m0 + sub) * NIN + half * 2;
    const float* a1p = a0p + 16 * NIN;
    const float* b0p = w + (size_t)(n0 + sub) * NIN + half * 2;
    const float* b1p = b0p + 16 * NIN;

    v8f c00 = {}, c01 = {}, c10 = {}, c11 = {};
    #pragma unroll 4
    for (int k = 0; k < NIN; k += 4) {
        v2f a0 = *(const v2f*)(a0p + k);
        v2f a1 = *(const v2f*)(a1p + k);
        v2f b0 = *(const v2f*)(b0p + k);
        v2f b1 = *(const v2f*)(b1p + k);
        c00 = WMMA_F32(a0, b0, c00);
        c01 = WMMA_F32(a0, b1, c01);
        c10 = WMMA_F32(a1, b0, c10);
        c11 = WMMA_F32(a1, b1, c11);
    }

    float* o = g0 + (size_t)(m0 + half * 8) * NOUT + n0 + sub;
    #pragma unroll
    for (int v = 0; v < 8; ++v) {
        o[(size_t)v * NOUT]              = c00[v];   // M tile 0, N tile 0
        o[(size_t)v * NOUT + 16]         = c01[v];   // M tile 0, N tile 1
        o[(size_t)(v + 16) * NOUT]       = c10[v];   // M tile 1, N tile 0
        o[(size_t)(v + 16) * NOUT + 16]  = c11[v];   // M tile 1, N tile 1
    }
}

// ---------------------------------------------------------------------------
// 21-step greedy pursuit, one block per batch row; all state in LDS.
// ---------------------------------------------------------------------------
__global__ __launch_bounds__(256) void bmp_iter(const float* __restrict__ g0,
                                                const float* __restrict__ w,
                                                const float* __restrict__ wt,
                                                const float* __restrict__ lam_p,
                                                float* __restrict__ enc_out,
                                                float* __restrict__ xr_out) {
    const int b = blockIdx.x;
    const int t = threadIdx.x;

    __shared__ float enc[NOUT];    // 8 KB
    __shared__ float yv[NIN];      // 2 KB
    __shared__ float rmax[256];
    __shared__ int   rmaxi[256];
    __shared__ int   idx6[KIN];
    __shared__ int   nsel;

    const float lambd = lam_p[0];
    const float* g0row = g0 + (size_t)b * NOUT;

    for (int o = t; o < NOUT; o += 256) enc[o] = 0.0f;
    for (int i = t; i < NIN;  i += 256) yv[i] = 0.0f;
    if (t == 0) nsel = 0;
    __syncthreads();

    for (int step = 0; step < KACT; ++step) {
        // ---- residual = 2*G0 - sum_{j} Wt[idx6[j], :] - lambd*enc ; argmax ----
        const int ns = nsel;
        float bestv = -__builtin_inff();
        int   besti = 0;
        for (int o = t; o < NOUT; o += 256) {
            float r = 2.0f * g0row[o] - lambd * enc[o];
            for (int j = 0; j < ns; ++j)
                r -= wt[idx6[j] * NOUT + o];
            if (r > bestv) { bestv = r; besti = o; }
        }
        rmax[t] = bestv; rmaxi[t] = besti;
        __syncthreads();
        for (int s = 128; s > 0; s >>= 1) {
            if (t < s) {
                float v2 = rmax[t + s]; int i2 = rmaxi[t + s];
                if (v2 > rmax[t] || (v2 == rmax[t] && i2 < rmaxi[t])) {
                    rmax[t] = v2; rmaxi[t] = i2;
                }
            }
            __syncthreads();
        }
        const int jb = rmaxi[0];
        if (t == 0) enc[jb] += 1.0f;

        // ---- y += W[jb, :] ----
        for (int i = t; i < NIN; i += 256) yv[i] += w[jb * NIN + i];
        __syncthreads();

        // ---- top-KIN of y (ties -> lowest index) ----
        for (int kk = 0; kk < KIN; ++kk) {
            float bv = -__builtin_inff();
            int   bi = 0;
            for (int i = t; i < NIN; i += 256) {
                bool taken = false;
                for (int j = 0; j < kk; ++j) taken |= (idx6[j] == i);
                float v = taken ? -__builtin_inff() : yv[i];
                if (v > bv) { bv = v; bi = i; }
            }
            rmax[t] = bv; rmaxi[t] = bi;
            __syncthreads();
            for (int s = 128; s > 0; s >>= 1) {
                if (t < s) {
                    float v2 = rmax[t + s]; int i2 = rmaxi[t + s];
                    if (v2 > rmax[t] || (v2 == rmax[t] && i2 < rmaxi[t])) {
                        rmax[t] = v2; rmaxi[t] = i2;
                    }
                }
                __syncthreads();
            }
            if (t == 0) idx6[kk] = rmaxi[0];
            __syncthreads();
        }
        if (t == 0) nsel = KIN;
        __syncthreads();
    }

    // ---- outputs: encoded row, then binary xr row ----
    for (int o = t; o < NOUT; o += 256) enc_out[(size_t)b * NOUT + o] = enc[o];
    for (int i = t; i < NIN;  i += 256) xr_out[(size_t)b * NIN + i] = 0.0f;
    __syncthreads();
    if (t < KIN) xr_out[(size_t)b * NIN + idx6[t]] = 1.0f;
}

// ---------------------------------------------------------------------------
extern "C" void kernel_launch(void* const* d_in, const int* in_sizes, int n_in,
                              void* d_out, int out_size, void* d_ws, size_t ws_size,
                              hipStream_t stream) {
    const float* x = (const float*)d_in[0];   // [512, 512]
    const float* w = (const float*)d_in[1];   // [2048, 512]

    float* out     = (float*)d_out;
    float* enc_out = out;                         // [512, 2048]
    float* xr_out  = out + (size_t)BATCH * NOUT;  // [512, 512]

    float* G0  = (float*)d_ws;                        // 512*2048 floats (4 MB)
    float* Wt  = G0 + (size_t)BATCH * NOUT;           // 512*2048 floats (4 MB)
    float* lam = Wt + (size_t)NIN * NOUT;             // 1 float

    lambd_kernel<<<1, 512, 0, stream>>>(x, lam);
    transpose_w<<<dim3(NIN / 32, NOUT / 32), 256, 0, stream>>>(w, Wt);
    gemm_xwT<<<dim3(NOUT / 128, BATCH / 64), 256, 0, stream>>>(x, w, G0);
    bmp_iter<<<BATCH, 256, 0, stream>>>(G0, w, Wt, lam, enc_out, xr_out);
}